// YarnRope_34376918237781
// MI455X (gfx1250) — compile-verified
//
#include <hip/hip_runtime.h>
#include <hip/hip_bf16.h>
#include <math.h>

// ---------------------------------------------------------------------------
// YaRN RoPE, fp32, HBM-bandwidth-bound (~640 MiB streamed => ~29us @ 23.3TB/s)
// CDNA5 path: GLOBAL_LOAD_ASYNC_TO_LDS_B128 staging (ASYNCcnt, NT hint) +
// per-token cos/sin table in LDS + 128-bit non-temporal output stores.
// ---------------------------------------------------------------------------

typedef float v4f __attribute__((ext_vector_type(4)));
typedef int   v4i __attribute__((ext_vector_type(4)));
typedef __attribute__((address_space(1))) v4i* g1_v4i;   // global int4*
typedef __attribute__((address_space(3))) v4i* l3_v4i;   // LDS int4*

#ifndef ASYNC_LOAD_CPOL
#define ASYNC_LOAD_CPOL 1   // TH[2:0]=1 -> TH_LOAD_NT (read-once stream)
#endif

__device__ __forceinline__ void async_copy_b128(const void* gsrc, void* ldst) {
#if __has_builtin(__builtin_amdgcn_global_load_async_to_lds_b128)
  __builtin_amdgcn_global_load_async_to_lds_b128((g1_v4i)(void*)gsrc, (l3_v4i)ldst,
                                                 /*offset=*/0, /*cpol=*/ASYNC_LOAD_CPOL);
#else
  unsigned loff = (unsigned)(unsigned long long)(__attribute__((address_space(3))) void*)ldst;
  unsigned long long gaddr = (unsigned long long)gsrc;
  asm volatile("global_load_async_to_lds_b128 %0, %1, off th:TH_LOAD_NT"
               :: "v"(loff), "v"(gaddr) : "memory");
#endif
}

__device__ __forceinline__ void wait_asynccnt0() {
#if __has_builtin(__builtin_amdgcn_s_wait_asynccnt)
  __builtin_amdgcn_s_wait_asynccnt(0);
#else
  asm volatile("s_wait_asynccnt 0" ::: "memory");
#endif
}

#define BLOCK_THREADS 320   // 10 wave32 waves; divides both loops exactly

__global__ __launch_bounds__(BLOCK_THREADS) void yarn_rope_kernel(
    const float* __restrict__ q, const float* __restrict__ k,
    const int* __restrict__ indptr, const int* __restrict__ offsets,
    float* __restrict__ outQ, float* __restrict__ outK, int nSeq)
{
  constexpr int HQ = 32, HK = 8, D = 128, HALF = 64;
  constexpr int QF = HQ * D;            // 4096 floats / token
  constexpr int KF = HK * D;            // 1024 floats / token
  __shared__ float lds[QF + KF];        // 20 KB token stage
  __shared__ float csTab[HALF];
  __shared__ float snTab[HALF];

  const int tok = blockIdx.x;
  const int tid = threadIdx.x;
  const float* qTok = q + (size_t)tok * QF;
  const float* kTok = k + (size_t)tok * KF;

  // ---- Stage q+k for this token into LDS via async DMA (no VGPR dests). ----
  // (QF+KF)/4 = 1280 float4 chunks == exactly 4 per thread at 320 threads.
  #pragma unroll
  for (int idx = tid; idx < (QF + KF) / 4; idx += BLOCK_THREADS) {
    const float* src = (idx < QF / 4) ? (qTok + idx * 4)
                                      : (kTok + (idx - QF / 4) * 4);
    async_copy_b128(src, &lds[idx * 4]);
  }

  // ---- Meanwhile, lanes 0..63 build the cos/sin table for this token. ----
  if (tid < HALF) {
    // seq_id = searchsorted(indptr, tok, 'right') - 1  (linear scan, 8 seqs)
    int sid = 0;
    for (int i = 1; i <= nSeq; ++i) sid = (tok >= indptr[i]) ? i : sid;
    const float pos = (float)(offsets[sid] + tok - indptr[sid]);

    const float jf = (float)tid;
    const float PI_F = 3.14159265358979323846f;
    const float THETA = 10000.0f, SCALE = 4.0f;
    const float ORIG = 16384.0f;
    // inv_extra = THETA^(-2j/128) = exp2(-(j/64)*log2(THETA))
    const float LOG2_THETA = 13.287712379549449f;
    const float inv_extra  = exp2f(-(jf * (1.0f / 64.0f)) * LOG2_THETA);
    const float inv_interp = inv_extra * (1.0f / SCALE);
    // ramp endpoints exactly per reference (constant-folds: LOW=30, HIGH=55)
    const float denom = 2.0f * logf(THETA);
    const float lowf  = fmaxf(floorf(128.0f * logf(ORIG / (32.0f * 2.0f * PI_F)) / denom), 0.0f);
    const float highf = fminf(ceilf (128.0f * logf(ORIG / ( 1.0f * 2.0f * PI_F)) / denom), 127.0f);
    const float ramp  = fminf(fmaxf((jf - lowf) / (highf - lowf), 0.0f), 1.0f);
    const float inv_freq = inv_interp * ramp + inv_extra * (1.0f - ramp);
    const float mscale = 0.1f * logf(SCALE) + 1.0f;

    float sv, cv;
    sincosf(pos * inv_freq, &sv, &cv);
    csTab[tid] = cv * mscale;
    snTab[tid] = sv * mscale;
  }

  wait_asynccnt0();     // DMA -> LDS complete for this wave
  __syncthreads();      // table + staged data visible to all 10 waves

  // ---- Rotate-half and stream out with non-temporal 128-bit stores. ----
  // 640 chunks == exactly 2 per thread: head h in [0,40), float4 pair
  // (x[j..j+3], x[j+64..j+67]).
  #pragma unroll
  for (int t = tid; t < (HQ + HK) * (HALF / 4); t += BLOCK_THREADS) {
    const int h  = t >> 4;
    const int c4 = (t & 15) * 4;
    const bool isQ = (h < HQ);
    const int ldsF = (isQ ? h * D : QF + (h - HQ) * D) + c4;

    const v4f a = *(const v4f*)&lds[ldsF];          // first half
    const v4f b = *(const v4f*)&lds[ldsF + HALF];   // second half
    const v4f C = *(const v4f*)&csTab[c4];
    const v4f S = *(const v4f*)&snTab[c4];

    const v4f o1 = a * C - b * S;
    const v4f o2 = b * C + a * S;

    float* outp = isQ ? (outQ + (size_t)tok * QF + h * D + c4)
                      : (outK + (size_t)tok * KF + (h - HQ) * D + c4);
    __builtin_nontemporal_store(o1, (v4f*)outp);
    __builtin_nontemporal_store(o2, (v4f*)(outp + HALF));
  }
}

extern "C" void kernel_launch(void* const* d_in, const int* in_sizes, int n_in,
                              void* d_out, int out_size, void* d_ws, size_t ws_size,
                              hipStream_t stream) {
  (void)n_in; (void)out_size; (void)d_ws; (void)ws_size;
  const float* q       = (const float*)d_in[0];
  const float* k       = (const float*)d_in[1];
  const int*   indptr  = (const int*)d_in[2];
  const int*   offsets = (const int*)d_in[3];

  const int tokens = in_sizes[0] / (32 * 128);   // 16384
  const int nSeq   = in_sizes[3];                // 8

  float* outQ = (float*)d_out;
  float* outK = outQ + (size_t)in_sizes[0];

  yarn_rope_kernel<<<tokens, BLOCK_THREADS, 0, stream>>>(q, k, indptr, offsets,
                                                         outQ, outK, nSeq);
}